// SetAbstractionLayer_38371237823018
// MI455X (gfx1250) — compile-verified
//
#include <hip/hip_runtime.h>
#include <math.h>

// ---------------------------------------------------------------------------
// Problem constants (match reference setup_inputs / reference())
// ---------------------------------------------------------------------------
constexpr int N_PTS   = 131072;
constexpr int M_CENT  = 2048;
constexpr int K_GRP   = 32;
constexpr int R_ROWS  = M_CENT * K_GRP;   // 65536
constexpr int NUM_FREQ = 10;
constexpr float PI_F  = 3.14159265358979323846f;

// FPS cooperative kernel geometry
constexpr int FPS_BLOCKS  = 64;
constexpr int FPS_THREADS = 256;
constexpr int FPS_CH      = N_PTS / FPS_BLOCKS;     // 2048 points per block
constexpr int FPS_PT      = FPS_CH / FPS_THREADS;   // 8 points per thread

typedef float v2f __attribute__((ext_vector_type(2)));
typedef float v8f __attribute__((ext_vector_type(8)));

// ---------------------------------------------------------------------------
// Grid-wide barrier (persistent cooperative kernel). bar[0]=count, bar[1]=gen.
// ---------------------------------------------------------------------------
__device__ __forceinline__ void grid_barrier(unsigned* bar) {
  __syncthreads();
  if (threadIdx.x == 0) {
    __threadfence();
    unsigned* cnt = bar;
    unsigned* gen = bar + 1;
    unsigned old = __hip_atomic_load(gen, __ATOMIC_RELAXED, __HIP_MEMORY_SCOPE_AGENT);
    unsigned prev = __hip_atomic_fetch_add(cnt, 1u, __ATOMIC_ACQ_REL, __HIP_MEMORY_SCOPE_AGENT);
    if (prev == (unsigned)(FPS_BLOCKS - 1)) {
      __hip_atomic_store(cnt, 0u, __ATOMIC_RELAXED, __HIP_MEMORY_SCOPE_AGENT);
      __hip_atomic_fetch_add(gen, 1u, __ATOMIC_ACQ_REL, __HIP_MEMORY_SCOPE_AGENT);
    } else {
      while (__hip_atomic_load(gen, __ATOMIC_ACQUIRE, __HIP_MEMORY_SCOPE_AGENT) == old) {
        __builtin_amdgcn_s_sleep(2);
      }
    }
    __threadfence();
  }
  __syncthreads();
}

__global__ void init_bar_kernel(unsigned* bar) { bar[0] = 0u; bar[1] = 0u; }

// ---------------------------------------------------------------------------
// Furthest point sampling: persistent kernel, dist/points in registers,
// one grid barrier per iteration (double-buffered per-block candidates).
// Tie-breaks pick the lowest index (jnp.argmax first-occurrence semantics).
// ---------------------------------------------------------------------------
__global__ __launch_bounds__(FPS_THREADS) void fps_kernel(
    const float* __restrict__ pts, float* __restrict__ cent,
    float* __restrict__ candv, int* __restrict__ candi, unsigned* __restrict__ bar) {
  __shared__ float sv[FPS_THREADS];
  __shared__ int   si[FPS_THREADS];
  __shared__ int   swin;

  const int t = threadIdx.x;
  const int b = blockIdx.x;
  const int base = b * FPS_CH + t;

  float px[FPS_PT], py[FPS_PT], d[FPS_PT];
#pragma unroll
  for (int j = 0; j < FPS_PT; ++j) {
    const float2 p = *reinterpret_cast<const float2*>(pts + 2 * (base + j * FPS_THREADS));
    px[j] = p.x; py[j] = p.y; d[j] = INFINITY;
  }

  float cx = pts[0], cy = pts[1];
  if (b == 0 && t == 0) { cent[0] = cx; cent[1] = cy; }

  // --- initial scan vs point 0, writes candidate buffer 0 ---
  {
    float bv = -1.0f; int bi = 0;
#pragma unroll
    for (int j = 0; j < FPS_PT; ++j) {
      const float dx = px[j] - cx, dy = py[j] - cy;
      const float nd = sqrtf(dx * dx + dy * dy);
      const float nv = fminf(d[j], nd);
      d[j] = nv;
      if (nv > bv) { bv = nv; bi = base + j * FPS_THREADS; }
    }
    sv[t] = bv; si[t] = bi; __syncthreads();
    for (int off = FPS_THREADS / 2; off > 0; off >>= 1) {
      if (t < off) {
        const float ov = sv[t + off]; const int oi = si[t + off];
        if (ov > sv[t] || (ov == sv[t] && oi < si[t])) { sv[t] = ov; si[t] = oi; }
      }
      __syncthreads();
    }
    if (t == 0) { candv[b] = sv[0]; candi[b] = si[0]; }
  }
  grid_barrier(bar);

  for (int s = 1; s < M_CENT; ++s) {
    // --- reduce previous iteration's candidates (redundantly per block) ---
    const int pb = (s - 1) & 1;
    if (t == 0) {
      float bv = candv[pb * FPS_BLOCKS]; int bi = candi[pb * FPS_BLOCKS];
      for (int k = 1; k < FPS_BLOCKS; ++k) {
        const float ov = candv[pb * FPS_BLOCKS + k]; const int oi = candi[pb * FPS_BLOCKS + k];
        if (ov > bv || (ov == bv && oi < bi)) { bv = ov; bi = oi; }
      }
      swin = bi;
    }
    __syncthreads();
    const int w = swin;
    const float2 cw = *reinterpret_cast<const float2*>(pts + 2 * w);
    cx = cw.x; cy = cw.y;
    if (b == 0 && t == 0) { cent[2 * s] = cx; cent[2 * s + 1] = cy; }

    // dist[w] = 0 (owner thread, static register indices only)
    {
      const int loc = w - b * FPS_CH;
      const bool own = (loc >= 0) && (loc < FPS_CH) && ((loc & (FPS_THREADS - 1)) == t);
      const int jz = loc >> 8;
#pragma unroll
      for (int j = 0; j < FPS_PT; ++j)
        if (own && j == jz) d[j] = 0.0f;
    }

    if (s < M_CENT - 1) {
      // --- scan vs new centroid, write candidate buffer (s&1) ---
      float bv = -1.0f; int bi = 0;
#pragma unroll
      for (int j = 0; j < FPS_PT; ++j) {
        const float dx = px[j] - cx, dy = py[j] - cy;
        const float nd = sqrtf(dx * dx + dy * dy);
        const float nv = fminf(d[j], nd);
        d[j] = nv;
        if (nv > bv) { bv = nv; bi = base + j * FPS_THREADS; }
      }
      sv[t] = bv; si[t] = bi; __syncthreads();
      for (int off = FPS_THREADS / 2; off > 0; off >>= 1) {
        if (t < off) {
          const float ov = sv[t + off]; const int oi = si[t + off];
          if (ov > sv[t] || (ov == sv[t] && oi < si[t])) { sv[t] = ov; si[t] = oi; }
        }
        __syncthreads();
      }
      const int wb = s & 1;
      if (t == 0) { candv[wb * FPS_BLOCKS + b] = sv[0]; candi[wb * FPS_BLOCKS + b] = si[0]; }
      grid_barrier(bar);
    }
  }
}

// ---------------------------------------------------------------------------
// Nearest-centroid assignment; centroids cached in LDS. argmin lowest-index.
// d2 = |p|^2 + |c|^2 - 2 p.c  (same formula as reference).
// ---------------------------------------------------------------------------
__global__ __launch_bounds__(256) void assign_kernel(
    const float* __restrict__ pts, const float* __restrict__ cent, int* __restrict__ assign) {
  __shared__ float cx[M_CENT], cy[M_CENT], c2[M_CENT];
  for (int m = threadIdx.x; m < M_CENT; m += 256) {
    const float a = cent[2 * m], bb = cent[2 * m + 1];
    cx[m] = a; cy[m] = bb; c2[m] = a * a + bb * bb;
  }
  __syncthreads();
  const int n = blockIdx.x * 256 + threadIdx.x;
  const float x = pts[2 * n], y = pts[2 * n + 1];
  const float p2 = x * x + y * y;
  float bd = INFINITY; int bm = 0;
  for (int m = 0; m < M_CENT; ++m) {
    const float d2 = (p2 + c2[m]) - 2.0f * (x * cx[m] + y * cy[m]);
    if (d2 < bd) { bd = d2; bm = m; }
  }
  assign[n] = bm;
}

// ---------------------------------------------------------------------------
// Select up to K=32 lowest-index points per centroid (wave32 ballot keeps
// ascending-index order). Pads with last selected; empty -> centroid/zeros.
// ---------------------------------------------------------------------------
__global__ __launch_bounds__(256) void select_kernel(
    const int* __restrict__ assign, const float* __restrict__ pts,
    const float* __restrict__ cent, float* __restrict__ gp) {
  __shared__ int sel[8][K_GRP];
  __shared__ int scnt[8];
  const int t = threadIdx.x;
  const int w = t >> 5;
  const int lane = t & 31;
  const int m = blockIdx.x * 8 + w;

  int cnt = 0;
  for (int base = 0; base < N_PTS; base += 32) {
    const int a = assign[base + lane];
    const bool pred = (a == m);
    const unsigned mask = (unsigned)__ballot(pred);
    if (pred) {
      const int pos = cnt + __popc(mask & ((1u << lane) - 1u));
      if (pos < K_GRP) sel[w][pos] = base + lane;
    }
    cnt += __popc(mask);
    if (cnt >= K_GRP) break;
  }
  if (lane == 0) scnt[w] = cnt;
  __syncthreads();

  const int cn = scnt[w];
  if (cn == 0) {
    if (lane == 0) {
      gp[(m * K_GRP) * 2]     = cent[2 * m];
      gp[(m * K_GRP) * 2 + 1] = cent[2 * m + 1];
    } else {
      gp[(m * K_GRP + lane) * 2]     = 0.0f;
      gp[(m * K_GRP + lane) * 2 + 1] = 0.0f;
    }
  } else {
    const int kk = min(lane, min(cn, K_GRP) - 1);
    const int idx = sel[w][kk];
    gp[(m * K_GRP + lane) * 2]     = pts[2 * idx];
    gp[(m * K_GRP + lane) * 2 + 1] = pts[2 * idx + 1];
  }
}

// ---------------------------------------------------------------------------
// High-frequency positional encoding: [M*K, 40], order f*4 + dim*2 + {sin,cos}
// ---------------------------------------------------------------------------
__global__ __launch_bounds__(256) void encode_kernel(
    const float* __restrict__ gp, const float* __restrict__ cent, float* __restrict__ enc) {
  const int i = blockIdx.x * 256 + threadIdx.x;  // 65536
  const int m = i >> 5;
  const float rx = gp[2 * i]     - cent[2 * m];
  const float ry = gp[2 * i + 1] - cent[2 * m + 1];
  float* e = enc + (size_t)i * (4 * NUM_FREQ);
#pragma unroll
  for (int k = 0; k < NUM_FREQ; ++k) {
    const float fr = PI_F * (float)(1 << k);  // exact power-of-two scaling
    const float ax = rx * fr, ay = ry * fr;
    e[4 * k + 0] = sinf(ax);
    e[4 * k + 1] = cosf(ax);
    e[4 * k + 2] = sinf(ay);
    e[4 * k + 3] = cosf(ay);
  }
}

// ---------------------------------------------------------------------------
// FP32 WMMA GEMM + bias: C[R,OUT] = A[R,IN] @ W[IN,OUT] + bias.
// One wave -> one 16x16 output tile, K-loop in steps of 4 via
// v_wmma_f32_16x16x4_f32. Fragment layouts per CDNA5 ISA §7.12.2:
//   A 16x4 f32: lanes 0-15 rows, VGPR v / half h -> K = k0 + v + 2h
//   B 4x16 f32: lanes hold N, VGPR v / half h -> K = k0 + v + 2h
//   C 16x16 f32: VGPR i / half h -> M = i + 8h, N = lane&15
// ---------------------------------------------------------------------------
template <int IN, int OUT>
__global__ __launch_bounds__(256) void gemm_bias_kernel(
    const float* __restrict__ A, const float* __restrict__ W,
    const float* __restrict__ bias, float* __restrict__ C) {
  const int lane  = threadIdx.x & 31;
  const int wave  = threadIdx.x >> 5;
  const int mtile = blockIdx.x * 8 + wave;
  const int ntile = blockIdx.y;
  const int nl    = lane & 15;
  const int half  = lane >> 4;
  const int row   = mtile * 16 + nl;   // A row for this lane
  const int col   = ntile * 16 + nl;   // B/C column for this lane

  v8f c;
  {
    const float bv = bias[col];
#pragma unroll
    for (int i = 0; i < 8; ++i) c[i] = bv;
  }

#pragma unroll
  for (int k0 = 0; k0 < IN; k0 += 4) {
    const int ka = k0 + 2 * half;
    const float2 av = *reinterpret_cast<const float2*>(A + (size_t)row * IN + ka);
    v2f a, b;
    a.x = av.x; a.y = av.y;
    b.x = W[(size_t)ka * OUT + col];
    b.y = W[(size_t)(ka + 1) * OUT + col];
    c = __builtin_amdgcn_wmma_f32_16x16x4_f32(false, a, false, b, (short)0, c, false, false);
  }

#pragma unroll
  for (int i = 0; i < 8; ++i) {
    C[(size_t)(mtile * 16 + half * 8 + i) * OUT + col] = c[i];
  }
}

// ---------------------------------------------------------------------------
// BatchNorm: deterministic two-level reduction (per-block partials, then a
// fixed-order final reduce in double). scale/shift folded:  y = sc*x + sh.
// ---------------------------------------------------------------------------
template <int CH>
__global__ __launch_bounds__(256) void bn_partial_kernel(
    const float* __restrict__ X, float* __restrict__ ps, float* __restrict__ pq) {
  constexpr int NG = 256 / CH;
  const int t = threadIdx.x;
  const int c = t & (CH - 1);
  const int g = t / CH;
  const int r0 = blockIdx.x * 256;  // 256 rows per block, 256 blocks
  float s = 0.0f, q = 0.0f;
  for (int r = g; r < 256; r += NG) {
    const float v = X[(size_t)(r0 + r) * CH + c];
    s += v; q += v * v;
  }
  __shared__ float ls[256], lq[256];
  ls[t] = s; lq[t] = q;
  __syncthreads();
  if (t < CH) {
    float ss = 0.0f, qq = 0.0f;
    for (int g2 = 0; g2 < NG; ++g2) { ss += ls[g2 * CH + t]; qq += lq[g2 * CH + t]; }
    ps[blockIdx.x * CH + t] = ss;
    pq[blockIdx.x * CH + t] = qq;
  }
}

template <int CH>
__global__ void bn_finalize_kernel(
    const float* __restrict__ ps, const float* __restrict__ pq,
    const float* __restrict__ gamma, const float* __restrict__ beta,
    float* __restrict__ scale, float* __restrict__ shift) {
  const int t = threadIdx.x;
  if (t >= CH) return;
  double s = 0.0, q = 0.0;
  for (int b = 0; b < 256; ++b) { s += (double)ps[b * CH + t]; q += (double)pq[b * CH + t]; }
  const double mean = s / (double)R_ROWS;
  double var = q / (double)R_ROWS - mean * mean;
  if (var < 0.0) var = 0.0;
  const float inv = (float)(1.0 / sqrt(var + 1e-5));
  const float sc = gamma[t] * inv;
  scale[t] = sc;
  shift[t] = beta[t] - sc * (float)mean;
}

template <int CH>
__global__ __launch_bounds__(256) void apply_bn_relu_kernel(
    float* __restrict__ X, const float* __restrict__ scale, const float* __restrict__ shift) {
  const size_t i = (size_t)blockIdx.x * 256 + threadIdx.x;
  const int c = (int)(i & (CH - 1));
  const float v = X[i] * scale[c] + shift[c];
  X[i] = fmaxf(v, 0.0f);
}

// ---------------------------------------------------------------------------
// Max-pool over K=32 per group + assemble d_out = [features(2048x16), centroids(2048x2)]
// ---------------------------------------------------------------------------
__global__ __launch_bounds__(256) void maxpool_out_kernel(
    const float* __restrict__ Y, const float* __restrict__ cent, float* __restrict__ out) {
  const int i = blockIdx.x * 256 + threadIdx.x;  // 36864 threads exactly
  if (i < M_CENT * 16) {
    const int m = i >> 4, c = i & 15;
    const float* p = Y + (size_t)(m << 5) * 16 + c;
    float mx = -INFINITY;
#pragma unroll
    for (int k = 0; k < K_GRP; ++k) mx = fmaxf(mx, p[k * 16]);
    out[i] = mx;
  } else {
    out[i] = cent[i - M_CENT * 16];
  }
}

// ---------------------------------------------------------------------------
// Host launcher
// ---------------------------------------------------------------------------
extern "C" void kernel_launch(void* const* d_in, const int* in_sizes, int n_in,
                              void* d_out, int out_size, void* d_ws, size_t ws_size,
                              hipStream_t stream) {
  (void)in_sizes; (void)n_in; (void)out_size; (void)ws_size;
  const float* pts = (const float*)d_in[0];
  const float* W1  = (const float*)d_in[1];
  const float* b1  = (const float*)d_in[2];
  const float* g1  = (const float*)d_in[3];
  const float* be1 = (const float*)d_in[4];
  const float* W2  = (const float*)d_in[5];
  const float* b2  = (const float*)d_in[6];
  const float* g2  = (const float*)d_in[7];
  const float* be2 = (const float*)d_in[8];
  const float* W3  = (const float*)d_in[9];
  const float* b3  = (const float*)d_in[10];

  char* base = (char*)d_ws;
  size_t off = 0;
  auto carve = [&](size_t bytes) -> void* {
    void* p = base + off;
    off = (off + bytes + 255) & ~(size_t)255;
    return p;
  };
  float*    cent  = (float*)carve((size_t)M_CENT * 2 * sizeof(float));
  float*    candv = (float*)carve(2 * FPS_BLOCKS * sizeof(float));
  int*      candi = (int*)carve(2 * FPS_BLOCKS * sizeof(int));
  unsigned* bar   = (unsigned*)carve(2 * sizeof(unsigned));
  int*      assgn = (int*)carve((size_t)N_PTS * sizeof(int));
  float*    gp    = (float*)carve((size_t)R_ROWS * 2 * sizeof(float));
  float*    enc   = (float*)carve((size_t)R_ROWS * 40 * sizeof(float));
  float*    h1    = (float*)carve((size_t)R_ROWS * 64 * sizeof(float));
  float*    h2    = (float*)carve((size_t)R_ROWS * 128 * sizeof(float));
  float*    h3    = (float*)carve((size_t)R_ROWS * 16 * sizeof(float));
  float*    ps    = (float*)carve((size_t)256 * 128 * sizeof(float));
  float*    pq    = (float*)carve((size_t)256 * 128 * sizeof(float));
  float*    scl   = (float*)carve(128 * sizeof(float));
  float*    shf   = (float*)carve(128 * sizeof(float));

  // 1) FPS (persistent cooperative kernel, 1 grid barrier per iteration)
  init_bar_kernel<<<1, 1, 0, stream>>>(bar);
  fps_kernel<<<FPS_BLOCKS, FPS_THREADS, 0, stream>>>(pts, cent, candv, candi, bar);

  // 2) Group: assign + ordered top-32 select + gather
  assign_kernel<<<N_PTS / 256, 256, 0, stream>>>(pts, cent, assgn);
  select_kernel<<<M_CENT / 8, 256, 0, stream>>>(assgn, pts, cent, gp);

  // 3) Encoding
  encode_kernel<<<R_ROWS / 256, 256, 0, stream>>>(gp, cent, enc);

  // 4) MLP layer 1: 40 -> 64, BN + ReLU
  gemm_bias_kernel<40, 64><<<dim3(R_ROWS / 128, 4), 256, 0, stream>>>(enc, W1, b1, h1);
  bn_partial_kernel<64><<<256, 256, 0, stream>>>(h1, ps, pq);
  bn_finalize_kernel<64><<<1, 64, 0, stream>>>(ps, pq, g1, be1, scl, shf);
  apply_bn_relu_kernel<64><<<(R_ROWS * 64) / 256, 256, 0, stream>>>(h1, scl, shf);

  // 5) MLP layer 2: 64 -> 128, BN + ReLU
  gemm_bias_kernel<64, 128><<<dim3(R_ROWS / 128, 8), 256, 0, stream>>>(h1, W2, b2, h2);
  bn_partial_kernel<128><<<256, 256, 0, stream>>>(h2, ps, pq);
  bn_finalize_kernel<128><<<1, 128, 0, stream>>>(ps, pq, g2, be2, scl, shf);
  apply_bn_relu_kernel<128><<<(R_ROWS * 128) / 256, 256, 0, stream>>>(h2, scl, shf);

  // 6) MLP layer 3: 128 -> 16 (bias only)
  gemm_bias_kernel<128, 16><<<dim3(R_ROWS / 128, 1), 256, 0, stream>>>(h2, W3, b3, h3);

  // 7) Max-pool over K + output assembly (features then centroids)
  maxpool_out_kernel<<<(M_CENT * 16 + M_CENT * 2) / 256, 256, 0, stream>>>(h3, cent, (float*)d_out);
}